// GFLayer_29197187678342
// MI455X (gfx1250) — compile-verified
//
#include <hip/hip_runtime.h>

// out[(o*2048+i)*9 + (h*3+w)] = sum_k weights[(o*2048+i)*3 + k] * x[k*9 + h*3+w]
// GEMM: M = 4194304 rows (16 per WMMA tile), K = 3 (pad to 4), N = 9 (pad to 16)
// Memory-bound: ~201 MB traffic -> ~8.6 us floor at 23.3 TB/s.
// fp32 precision required -> V_WMMA_F32_16X16X4_F32 (free compute on the matrix pipe).

#define M_TOTAL (2048 * 2048)
#define N_TILES (M_TOTAL / 16)   // 262144 tiles of 16 rows
#define N_PAIRS (N_TILES / 2)    // 131072 tile-pairs

typedef __attribute__((ext_vector_type(2))) float v2f;
typedef __attribute__((ext_vector_type(8))) float v8f;
// 4-byte-aligned float2 so a dword-aligned global_load_b64 is legal.
typedef __attribute__((ext_vector_type(2), aligned(4))) float v2f_a4;

// Build the 16x4 A tile for one 16-row block starting at wt (48 contiguous floats).
// A layout (wave32): VGPR0 = K0 (lanes 0-15) / K2 (lanes 16-31)
//                    VGPR1 = K1 (lanes 0-15) / K3 = 0 (lanes 16-31)
// Lanes 0-15 load b64 at 3r   -> (K0, K1)
// Lanes 16-31 load b64 at 3r+1 -> (K1, K2); select .y as K2, zero-pad K3.
// Max dword index touched: 3*15+2 = 47 < 48, so no overread past the tile.
__device__ __forceinline__ v2f load_a_tile(const float* __restrict__ wt,
                                           int r, int half) {
    v2f_a4 p = *(const v2f_a4*)(wt + 3 * r + half);
    v2f a;
    a.x = half ? p.y : p.x;
    a.y = half ? 0.0f : p.y;
    return a;
}

__global__ __launch_bounds__(256) void einsum_oik_khw_wmma(
    const float* __restrict__ w,    // [M_TOTAL][3]  (weights)
    const float* __restrict__ x,    // [3][9]        (basis filters)
    float* __restrict__ out)        // [M_TOTAL][9]
{
    const int lane   = threadIdx.x & 31;
    const int half   = lane >> 4;        // 0: lanes 0-15, 1: lanes 16-31
    const int r      = lane & 15;        // A: row-in-tile, B/C: column
    const int wave   = blockIdx.x * (blockDim.x >> 5) + (threadIdx.x >> 5);
    const int nwaves = gridDim.x * (blockDim.x >> 5);

    // ---- B matrix 4x16 (K x N), built once per wave ----
    // VGPR0: lanes 0-15 = K=0 row, lanes 16-31 = K=2 row
    // VGPR1: lanes 0-15 = K=1 row, lanes 16-31 = K=3 row (zero pad)
    v2f b;
    b.x = (r < 9) ? x[(half ? 2 : 0) * 9 + r] : 0.0f;
    b.y = (!half && r < 9) ? x[1 * 9 + r] : 0.0f;

    for (int tp = wave; tp < N_PAIRS; tp += nwaves) {
        const int t0 = tp * 2;                        // first tile of the pair
        const float* wt0 = w + t0 * 48;               // 96 contiguous floats (2 tiles)

        // Prefetch this wave's next pair (global_prefetch_b8; speculative).
        __builtin_prefetch(wt0 + 96 * nwaves, 0, 1);

        // ---- Two independent A loads + WMMAs (ILP, uniform control flow) ----
        v2f a0 = load_a_tile(wt0,      r, half);
        v2f a1 = load_a_tile(wt0 + 48, r, half);

        v8f z = {};
        v8f c0 = __builtin_amdgcn_wmma_f32_16x16x4_f32(
                     false, a0, false, b, (short)0, z, false, false);
        v8f c1 = __builtin_amdgcn_wmma_f32_16x16x4_f32(
                     false, a1, false, b, (short)0, z, false, false);

        // ---- Store the two 16x9 valid sub-blocks (1152 contiguous bytes) ----
        // VGPR j: lanes 0-15 hold C[j][r], lanes 16-31 hold C[j+8][r]
        float* ot = out + t0 * 144;                   // 16 rows * 9 per tile
        if (r < 9) {
            const int mrow = 8 * half;
            #pragma unroll
            for (int j = 0; j < 8; ++j)
                ot[(mrow + j) * 9 + r] = c0[j];
            #pragma unroll
            for (int j = 0; j < 8; ++j)
                ot[144 + (mrow + j) * 9 + r] = c1[j];
        }
    }
}

extern "C" void kernel_launch(void* const* d_in, const int* in_sizes, int n_in,
                              void* d_out, int out_size, void* d_ws, size_t ws_size,
                              hipStream_t stream) {
    // setup_inputs order: d_in[0] = x (3*3*3 f32), d_in[1] = weights (2048*2048*3 f32)
    const float* x = (const float*)d_in[0];
    const float* w = (const float*)d_in[1];
    float* out = (float*)d_out;

    dim3 block(256);   // 8 wave32s
    dim3 grid(8192);   // 65536 waves, 2 tile-pairs each over 131072 pairs
    hipLaunchKernelGGL(einsum_oik_khw_wmma, grid, block, 0, stream, w, x, out);
}